// GINMultiHeadEHM_3023656976410
// MI455X (gfx1250) — compile-verified
//
#include <hip/hip_runtime.h>

#define N_NODES  100000
#define N_EDGES  1600000
#define N_GRAPHS 512
#define HID      64
#define N_LAYERS 3
#define N_HEADS  20

typedef __attribute__((ext_vector_type(2))) float v2f;
typedef __attribute__((ext_vector_type(8))) float v8f;

// ---------------------------------------------------------------- utilities
__global__ void zero_kernel(float* __restrict__ p, int n) {
  int i = blockIdx.x * blockDim.x + threadIdx.x;
  if (i < n) p[i] = 0.0f;
}

// h[n,d] = embed[x[n], d]
__global__ void embed_kernel(const int* __restrict__ x,
                             const float* __restrict__ embed,
                             float* __restrict__ h) {
  int i = blockIdx.x * blockDim.x + threadIdx.x;
  if (i >= N_NODES * HID) return;
  int node = i >> 6, d = i & 63;
  h[i] = embed[(size_t)x[node] * HID + d];
}

// z = (1 + eps[layer]) * h
__global__ void scale_kernel(const float* __restrict__ h,
                             const float* __restrict__ eps, int layer,
                             float* __restrict__ z) {
  int i = blockIdx.x * blockDim.x + threadIdx.x;
  if (i >= N_NODES * HID) return;
  z[i] = (1.0f + eps[layer]) * h[i];
}

// z[dst[e], :] += h[src[e], :]   (64 lanes per edge, coalesced 256B groups)
__global__ void scatter_kernel(const int* __restrict__ src,
                               const int* __restrict__ dst,
                               const float* __restrict__ h,
                               float* __restrict__ z) {
  long long t = (long long)blockIdx.x * blockDim.x + threadIdx.x;
  int e = (int)(t >> 6), d = (int)(t & 63);
  if (e >= N_EDGES) return;
  atomicAdd(&z[(size_t)dst[e] * HID + d], h[(size_t)src[e] * HID + d]);
}

// ------------------------------------------------------- WMMA helper (f32)
// One wave computes a 16x64 output block: 4 N-tiles x 16 chained K=4 WMMAs.
// A-layout (V_WMMA_F32_16X16X4_F32): lanes 0-15 hold K={0,1}, lanes 16-31 K={2,3}.
__device__ __forceinline__ void wmma_16x64(const v2f* __restrict__ A,    // A[16] k-steps
                                           const float* __restrict__ B,  // [64,64] row-major
                                           int hi, int lo, v8f out[4]) {
#pragma unroll
  for (int nt = 0; nt < 4; ++nt) {
    v8f acc = {};
#pragma unroll
    for (int kk = 0; kk < 16; ++kk) {
      int krow = 4 * kk + 2 * hi;
      v2f b;
      b[0] = B[(size_t)krow * HID + nt * 16 + lo];
      b[1] = B[(size_t)(krow + 1) * HID + nt * 16 + lo];
      acc = __builtin_amdgcn_wmma_f32_16x16x4_f32(
          /*neg_a=*/false, A[kk], /*neg_b=*/false, b,
          /*c_mod=*/(short)0, acc, /*reuse_a=*/false, /*reuse_b=*/false);
    }
    out[nt] = acc;
  }
}

// -------------------------------------------- fused node MLP (per-layer)
// h_out = relu( relu(z @ W1 + b1) @ W2 + b2 ), one wave per 16-node tile.
__global__ __launch_bounds__(32) void gin_mlp_kernel(
    const float* __restrict__ z, const float* __restrict__ W1,
    const float* __restrict__ b1, const float* __restrict__ W2,
    const float* __restrict__ b2, float* __restrict__ h_out) {
  __shared__ float lds[16 * HID];  // one wave per block: private 4KB
  int row0 = blockIdx.x * 16;
  int lane = threadIdx.x;
  int hi = lane >> 4, lo = lane & 15;

  // ---- load A (z rows) in WMMA A-layout
  v2f A[16];
  const float* zrow = z + (size_t)(row0 + lo) * HID + 2 * hi;
#pragma unroll
  for (int kk = 0; kk < 16; ++kk) A[kk] = *(const v2f*)(zrow + 4 * kk);

  // ---- GEMM1 + bias + relu -> LDS (row-major 16x64)
  v8f t1[4];
  wmma_16x64(A, W1, hi, lo, t1);
#pragma unroll
  for (int nt = 0; nt < 4; ++nt) {
    int col = nt * 16 + lo;
    float bias = b1[col];
#pragma unroll
    for (int i = 0; i < 8; ++i) {
      float v = t1[nt][i] + bias;
      lds[(i + 8 * hi) * HID + col] = v > 0.0f ? v : 0.0f;
    }
  }

  // ---- re-swizzle: D-layout (via LDS) -> A-layout for GEMM2
  v2f A2[16];
#pragma unroll
  for (int kk = 0; kk < 16; ++kk)
    A2[kk] = *(const v2f*)(&lds[lo * HID + 4 * kk + 2 * hi]);

  // ---- GEMM2 + bias + relu -> h_out
  v8f t2[4];
  wmma_16x64(A2, W2, hi, lo, t2);
#pragma unroll
  for (int nt = 0; nt < 4; ++nt) {
    int col = nt * 16 + lo;
    float bias = b2[col];
#pragma unroll
    for (int i = 0; i < 8; ++i) {
      float v = t2[nt][i] + bias;
      h_out[(size_t)(row0 + i + 8 * hi) * HID + col] = v > 0.0f ? v : 0.0f;
    }
  }
}

// -------------------------------------------------------------- pooling
__global__ void pool_kernel(const float* __restrict__ h,
                            const int* __restrict__ batch,
                            float* __restrict__ sums, float* __restrict__ cnt) {
  int t = blockIdx.x * blockDim.x + threadIdx.x;
  int node = t >> 6, d = t & 63;
  if (node >= N_NODES) return;
  int b = batch[node];
  atomicAdd(&sums[(size_t)b * HID + d], h[(size_t)node * HID + d]);
  if (d == 0) atomicAdd(&cnt[b], 1.0f);
}

__global__ void gmean_kernel(const float* __restrict__ sums,
                             const float* __restrict__ cnt,
                             float* __restrict__ g) {
  int i = blockIdx.x * blockDim.x + threadIdx.x;
  if (i >= N_GRAPHS * HID) return;
  float c = cnt[i >> 6];
  g[i] = sums[i] / (c > 1.0f ? c : 1.0f);
}

// ------------------------------------- per-head GEMM: t = relu(g@hW1[k]+hb1[k])
__global__ __launch_bounds__(32) void head_gemm_kernel(
    const float* __restrict__ g, const float* __restrict__ hW1,
    const float* __restrict__ hb1, float* __restrict__ t_ws) {
  int head = blockIdx.x >> 5;      // 32 tiles per head (512/16)
  int tile = blockIdx.x & 31;
  int row0 = tile * 16;
  int lane = threadIdx.x;
  int hi = lane >> 4, lo = lane & 15;

  v2f A[16];
  const float* grow = g + (size_t)(row0 + lo) * HID + 2 * hi;
#pragma unroll
  for (int kk = 0; kk < 16; ++kk) A[kk] = *(const v2f*)(grow + 4 * kk);

  v8f t[4];
  wmma_16x64(A, hW1 + (size_t)head * HID * HID, hi, lo, t);

  float* tout = t_ws + (size_t)head * N_GRAPHS * HID;
#pragma unroll
  for (int nt = 0; nt < 4; ++nt) {
    int col = nt * 16 + lo;
    float bias = hb1[head * HID + col];
#pragma unroll
    for (int i = 0; i < 8; ++i) {
      float v = t[nt][i] + bias;
      tout[(size_t)(row0 + i + 8 * hi) * HID + col] = v > 0.0f ? v : 0.0f;
    }
  }
}

// out[g,k] = t[k,g,:] . hW2[k,:] + hb2[k]   (output already transposed)
__global__ void head_out_kernel(const float* __restrict__ t_ws,
                                const float* __restrict__ hW2,
                                const float* __restrict__ hb2,
                                float* __restrict__ out) {
  int idx = blockIdx.x * blockDim.x + threadIdx.x;
  if (idx >= N_GRAPHS * N_HEADS) return;
  int k = idx % N_HEADS, gi = idx / N_HEADS;
  const float* t = t_ws + ((size_t)k * N_GRAPHS + gi) * HID;
  const float* w = hW2 + (size_t)k * HID;
  float s = hb2[k];
#pragma unroll
  for (int d = 0; d < HID; ++d) s += t[d] * w[d];
  out[gi * N_HEADS + k] = s;
}

// ---------------------------------------------------------------- driver
extern "C" void kernel_launch(void* const* d_in, const int* in_sizes, int n_in,
                              void* d_out, int out_size, void* d_ws, size_t ws_size,
                              hipStream_t stream) {
  const int*   x     = (const int*)d_in[0];
  const int*   ei    = (const int*)d_in[1];   // [2, N_EDGES]
  const int*   batch = (const int*)d_in[2];
  const float* embed = (const float*)d_in[3];
  const float* eps   = (const float*)d_in[4];
  const float* W1    = (const float*)d_in[5];
  const float* b1    = (const float*)d_in[6];
  const float* W2    = (const float*)d_in[7];
  const float* b2    = (const float*)d_in[8];
  const float* hW1   = (const float*)d_in[9];
  const float* hb1   = (const float*)d_in[10];
  const float* hW2   = (const float*)d_in[11];
  const float* hb2   = (const float*)d_in[12];
  float* out = (float*)d_out;

  float* ws = (float*)d_ws;
  float* h = ws;                              // N_NODES*HID
  float* z = ws + (size_t)N_NODES * HID;      // N_NODES*HID (reused post-layers)
  float* sums = z;                            // N_GRAPHS*HID
  float* cnt  = sums + N_GRAPHS * HID;        // N_GRAPHS
  float* gbuf = cnt + N_GRAPHS;               // N_GRAPHS*HID
  float* t_ws = gbuf + N_GRAPHS * HID;        // N_HEADS*N_GRAPHS*HID

  const int* src = ei;
  const int* dst = ei + N_EDGES;

  const int nh_elems = N_NODES * HID;                 // 6.4M
  const int blk = 256;

  embed_kernel<<<(nh_elems + blk - 1) / blk, blk, 0, stream>>>(x, embed, h);

  for (int l = 0; l < N_LAYERS; ++l) {
    scale_kernel<<<(nh_elems + blk - 1) / blk, blk, 0, stream>>>(h, eps, l, z);
    long long et = (long long)N_EDGES * HID;
    scatter_kernel<<<(unsigned)((et + blk - 1) / blk), blk, 0, stream>>>(src, dst, h, z);
    gin_mlp_kernel<<<N_NODES / 16, 32, 0, stream>>>(
        z, W1 + (size_t)l * HID * HID, b1 + (size_t)l * HID,
        W2 + (size_t)l * HID * HID, b2 + (size_t)l * HID, h);
  }

  int pool_init = N_GRAPHS * HID + N_GRAPHS;
  zero_kernel<<<(pool_init + blk - 1) / blk, blk, 0, stream>>>(sums, pool_init);
  pool_kernel<<<(nh_elems + blk - 1) / blk, blk, 0, stream>>>(h, batch, sums, cnt);
  gmean_kernel<<<(N_GRAPHS * HID + blk - 1) / blk, blk, 0, stream>>>(sums, cnt, gbuf);

  head_gemm_kernel<<<N_HEADS * (N_GRAPHS / 16), 32, 0, stream>>>(gbuf, hW1, hb1, t_ws);
  head_out_kernel<<<(N_GRAPHS * N_HEADS + blk - 1) / blk, blk, 0, stream>>>(
      t_ws, hW2, hb2, out);
}